// ProjE_4544075399311
// MI455X (gfx1250) — compile-verified
//
#include <hip/hip_runtime.h>
#include <hip/hip_bf16.h>

typedef __attribute__((ext_vector_type(2))) float v2f;
typedef __attribute__((ext_vector_type(8))) float v8f;

#define EMB   128
#define MTILE 32    // batch rows per block (two 16-row WMMA sub-tiles)
#define NWAVE 8     // 8 wave32 per block, each owns a 16-wide N slice

__global__ __launch_bounds__(256) void proje_score_kernel(
    const int*   __restrict__ triple,        // (BATCH, 3) int32
    const float* __restrict__ entity_emb,    // (N_ENTITY, 128)
    const float* __restrict__ relation_emb,  // (N_RELATION, 128)
    const float* __restrict__ De,            // (128, 128)
    const float* __restrict__ Dr,            // (128, 128)
    const float* __restrict__ b_c,           // (BATCH, 128)
    float*       __restrict__ out)           // (BATCH,) == (BATCH,1)
{
    __shared__ float Hs[MTILE][EMB];   // gathered head rows
    __shared__ float Rs[MTILE][EMB];   // gathered relation rows
    __shared__ float score[MTILE];     // per-row dot accumulators
    __shared__ int   tIdx[MTILE];      // tail entity indices

    const int tid  = threadIdx.x;      // 0..255
    const int wave = tid >> 5;         // 0..7
    const int lane = tid & 31;
    const int hi   = lane >> 4;        // lane group (0: lanes 0-15, 1: 16-31)
    const int l16  = lane & 15;
    const int m0   = blockIdx.x * MTILE;

    if (tid < MTILE) {
        score[tid] = 0.0f;
        tIdx[tid]  = triple[(m0 + tid) * 3 + 2];
    }

    // ---- cooperative gather: 32 rows x 128 f32 of H and R into LDS ----
    // 256 threads -> 8 threads per row, 16 floats (4x float4) per thread.
    {
        const int row = tid >> 3;
        const int col = (tid & 7) * 16;
        const int b   = m0 + row;
        const int hidx = triple[b * 3 + 0];
        const int ridx = triple[b * 3 + 1];
        const float4* hsrc = (const float4*)(entity_emb   + (size_t)hidx * EMB + col);
        const float4* rsrc = (const float4*)(relation_emb + (size_t)ridx * EMB + col);
        float4 h0 = hsrc[0], h1 = hsrc[1], h2 = hsrc[2], h3 = hsrc[3];
        float4 r0 = rsrc[0], r1 = rsrc[1], r2 = rsrc[2], r3 = rsrc[3];
        *(float4*)&Hs[row][col]      = h0;
        *(float4*)&Hs[row][col + 4]  = h1;
        *(float4*)&Hs[row][col + 8]  = h2;
        *(float4*)&Hs[row][col + 12] = h3;
        *(float4*)&Rs[row][col]      = r0;
        *(float4*)&Rs[row][col + 4]  = r1;
        *(float4*)&Rs[row][col + 8]  = r2;
        *(float4*)&Rs[row][col + 12] = r3;
    }
    __syncthreads();

    // ---- WMMA: two 16x16 C tiles per wave sharing the same B fragments ----
    const int n0   = wave * 16;
    const int ncol = n0 + l16;

    v8f c0 = {};   // rows m0 .. m0+15
    v8f c1 = {};   // rows m0+16 .. m0+31
    #pragma unroll
    for (int k = 0; k < EMB; k += 4) {
        const int ka = k + 2 * hi;     // A: lanes 0-15 hold K={k,k+1}, lanes 16-31 K={k+2,k+3}
        v2f ah0, ah1, ar0, ar1, bD, bR;
        ah0[0] = Hs[l16][ka];          ah0[1] = Hs[l16][ka + 1];
        ah1[0] = Hs[l16 + 16][ka];     ah1[1] = Hs[l16 + 16][ka + 1];
        ar0[0] = Rs[l16][ka];          ar0[1] = Rs[l16][ka + 1];
        ar1[0] = Rs[l16 + 16][ka];     ar1[1] = Rs[l16 + 16][ka + 1];
        // B (4x16): row k striped across lanes; VGPR0 K={k | k+2}, VGPR1 K={k+1 | k+3}
        bD[0] = De[(size_t)ka * EMB + ncol];
        bD[1] = De[(size_t)(ka + 1) * EMB + ncol];
        bR[0] = Dr[(size_t)ka * EMB + ncol];
        bR[1] = Dr[(size_t)(ka + 1) * EMB + ncol];
        c0 = __builtin_amdgcn_wmma_f32_16x16x4_f32(false, ah0, false, bD, (short)0, c0, false, false);
        c1 = __builtin_amdgcn_wmma_f32_16x16x4_f32(false, ah1, false, bD, (short)0, c1, false, false);
        c0 = __builtin_amdgcn_wmma_f32_16x16x4_f32(false, ar0, false, bR, (short)0, c0, false, false);
        c1 = __builtin_amdgcn_wmma_f32_16x16x4_f32(false, ar1, false, bR, (short)0, c1, false, false);
    }

    // ---- epilogue: f = tanh(C + b_c); partial dot with gathered t; reduce ----
    #pragma unroll
    for (int sub = 0; sub < 2; ++sub) {
        #pragma unroll
        for (int v = 0; v < 8; ++v) {
            const int mrow = sub * 16 + v + 8 * hi;  // C layout: VGPR v, lanes 0-15 -> M=v, 16-31 -> M=v+8
            const float cv = (sub == 0) ? c0[v] : c1[v];
            float x = cv + b_c[(size_t)(m0 + mrow) * EMB + ncol];
            float f = tanhf(x);
            float tval = entity_emb[(size_t)tIdx[mrow] * EMB + ncol];
            float p = f * tval;
            // sum over the 16 lanes of this wave's column slice (stays within 16-lane halves)
            p += __shfl_xor(p, 1, 32);
            p += __shfl_xor(p, 2, 32);
            p += __shfl_xor(p, 4, 32);
            p += __shfl_xor(p, 8, 32);
            if (l16 == 0) atomicAdd(&score[mrow], p);   // ds_add_f32 across the 8 waves
        }
    }
    __syncthreads();

    if (tid < MTILE) {
        float s = score[tid];
        out[m0 + tid] = 1.0f / (1.0f + __expf(-s));
    }
}

extern "C" void kernel_launch(void* const* d_in, const int* in_sizes, int n_in,
                              void* d_out, int out_size, void* d_ws, size_t ws_size,
                              hipStream_t stream) {
    const int*   triple       = (const int*)  d_in[0];
    const float* entity_emb   = (const float*)d_in[1];
    const float* relation_emb = (const float*)d_in[2];
    const float* De           = (const float*)d_in[3];
    const float* Dr           = (const float*)d_in[4];
    const float* b_c          = (const float*)d_in[5];
    float* out = (float*)d_out;

    const int batch  = in_sizes[0] / 3;         // triple is (BATCH, 3)
    const int blocks = batch / MTILE;           // 131072 / 32 = 4096

    proje_score_kernel<<<blocks, 256, 0, stream>>>(
        triple, entity_emb, relation_emb, De, Dr, b_c, out);
}